// MultiScaleGlobalRouter_17428977287677
// MI455X (gfx1250) — compile-verified
//
#include <hip/hip_runtime.h>
#include <math.h>
#include <stdint.h>

typedef __attribute__((ext_vector_type(2))) float v2f;
typedef __attribute__((ext_vector_type(8))) float v8f;

// Problem sizes (fixed by the reference)
constexpr int D      = 2048;
constexpr int NTOK   = 4 * 4096;        // 16384 tokens
constexpr int NPAD   = 16;              // 11 weight rows padded to 16 (WMMA N)
constexpr int BK     = 64;              // K-chunk staged in LDS (double-buffered)
constexpr int NCHUNK = D / BK;          // 32
constexpr int TPB    = 128;             // 4 waves per block
constexpr int TOKB   = 64;              // tokens per block (4 waves x 16)
constexpr int NBLK   = NTOK / TOKB;     // 256 blocks
constexpr int LDX    = BK + 4;          // padded LDS row stride -> conflict-free frag reads
constexpr int SXHALF = TOKB * LDX;      // floats per x-tile buffer
constexpr int SWHALF = NPAD * LDX;      // floats per W'-tile buffer
constexpr size_t HEX_BASE = (size_t)NTOK * 3;
constexpr size_t LB_IDX   = (size_t)NTOK * 3 + (size_t)NTOK * 64;

// --- CDNA5 async global->LDS copy (ASYNCcnt path), inline asm -----------------
// LDS destination offset = low 32 bits of the generic shared-aperture pointer
// (ISA 10.2).
__device__ __forceinline__ uint32_t lds_u32(const void* p) {
  return (uint32_t)(uintptr_t)p;
}
__device__ __forceinline__ void async_ld_b128(uint32_t ldsoff, const float* g) {
  asm volatile("global_load_async_to_lds_b128 %0, %1, off"
               :: "v"(ldsoff), "v"(g) : "memory");
}
__device__ __forceinline__ void wait_async0() {
  asm volatile("s_wait_asynccnt 0x0" ::: "memory");
}

// ---------------------------------------------------------------------------
// Kernel 1: build W' = ln_w ⊙ [w_q2; w_q3; w_q6; 0pad]  (16 x 2048),
// S1[n] = Σ ln_w*w_n, S2[n] = Σ ln_b*w_n, plus softmaxed scalar params.
// params: [0:16) S1, [16:32) S2, [32] temp, [33] alpha, [34:37) mix, [40:64) q3g(8x3)
// ---------------------------------------------------------------------------
__global__ void prep_kernel(const float* __restrict__ ln_w, const float* __restrict__ ln_b,
                            const float* __restrict__ w2, const float* __restrict__ w3,
                            const float* __restrict__ w6, const float* __restrict__ log_temp,
                            const float* __restrict__ log_scale_mix,
                            const float* __restrict__ q3_to_group,
                            const float* __restrict__ log_wht_mix,
                            float* __restrict__ Wp, float* __restrict__ params) {
  const int tid = threadIdx.x;
  for (int idx = tid; idx < NPAD * D; idx += blockDim.x) {
    const int r = idx / D, c = idx - r * D;
    float w = 0.f;
    if (r < 2)       w = w2[r * D + c];
    else if (r < 5)  w = w3[(r - 2) * D + c];
    else if (r < 11) w = w6[(r - 5) * D + c];
    Wp[idx] = w * ln_w[c];
  }
  if (tid < NPAD) {
    float s1 = 0.f, s2 = 0.f;
    for (int c = 0; c < D; ++c) {
      float w = 0.f;
      if (tid < 2)       w = w2[tid * D + c];
      else if (tid < 5)  w = w3[(tid - 2) * D + c];
      else if (tid < 11) w = w6[(tid - 5) * D + c];
      s1 += w * ln_w[c];
      s2 += w * ln_b[c];
    }
    params[tid]      = s1;
    params[16 + tid] = s2;
  } else if (tid == 32) {
    float t = expf(log_temp[0]);
    params[32] = fminf(fmaxf(t, 0.1f), 5.0f);
    params[33] = 1.0f / (1.0f + expf(-log_wht_mix[0]));
    const float m0 = log_scale_mix[0], m1 = log_scale_mix[1], m2 = log_scale_mix[2];
    const float mx = fmaxf(m0, fmaxf(m1, m2));
    const float e0 = expf(m0 - mx), e1 = expf(m1 - mx), e2 = expf(m2 - mx);
    const float inv = 1.0f / (e0 + e1 + e2);
    params[34] = e0 * inv; params[35] = e1 * inv; params[36] = e2 * inv;
  } else if (tid >= 40 && tid < 48) {
    const int v = tid - 40;
    const float a = q3_to_group[v * 3 + 0], b = q3_to_group[v * 3 + 1], c = q3_to_group[v * 3 + 2];
    const float mx = fmaxf(a, fmaxf(b, c));
    const float ea = expf(a - mx), eb = expf(b - mx), ec = expf(c - mx);
    const float inv = 1.0f / (ea + eb + ec);
    params[40 + v * 3 + 0] = ea * inv;
    params[40 + v * 3 + 1] = eb * inv;
    params[40 + v * 3 + 2] = ec * inv;
  }
}

// ---------------------------------------------------------------------------
// Kernel 2: fused LN-stats + f32 WMMA projection + router epilogue.
// Double-buffered async global->LDS pipeline: issue chunk c+1, compute chunk c.
// ---------------------------------------------------------------------------
__global__ __launch_bounds__(TPB) void router_main(const float* __restrict__ x,
                                                   const float* __restrict__ Wp,
                                                   const float* __restrict__ params,
                                                   float* __restrict__ out,
                                                   float* __restrict__ part) {
  __shared__ float sx2[2 * SXHALF];   // x tiles (2 bufs); reused: shex(64x64), slog(64x16)
  __shared__ float sw2[2 * SWHALF];   // W' tiles (2 bufs); reused: sgw(64x3)
  __shared__ float spsum[TPB];
  __shared__ float spsq[TPB];
  __shared__ float smu[TOKB];
  __shared__ float srstd[TOKB];

  const int tid  = threadIdx.x;
  const int blk  = blockIdx.x;
  const int wave = tid >> 5;
  const int lane = tid & 31;
  const int tokBase = blk * TOKB;

  // Issue one K-chunk of async global->LDS copies (x: 64x16 float4, W': 16x16 float4)
  auto issue_chunk = [&](int c, int buf) {
    const int kc = c * BK;
    float* sxd = sx2 + buf * SXHALF;
    float* swd = sw2 + buf * SWHALF;
    #pragma unroll
    for (int i = 0; i < 8; ++i) {
      const int idx = i * TPB + tid;
      const int row = idx >> 4;          // 16 float4 per row
      const int c4  = idx & 15;
      async_ld_b128(lds_u32(&sxd[row * LDX + c4 * 4]),
                    x + (size_t)(tokBase + row) * D + kc + c4 * 4);
    }
    #pragma unroll
    for (int i = 0; i < 2; ++i) {
      const int idx = i * TPB + tid;
      const int row = idx >> 4;
      const int c4  = idx & 15;
      async_ld_b128(lds_u32(&swd[row * LDX + c4 * 4]),
                    Wp + row * D + kc + c4 * 4);
    }
  };

  v8f acc = {};
  float psum = 0.f, psq = 0.f;
  int cur = 0;

  issue_chunk(0, 0);

  for (int c = 0; c < NCHUNK; ++c) {
    wait_async0();        // chunk c resident in buf[cur]
    __syncthreads();      // all waves' copies done; prior readers of buf[cur^1] done
    if (c + 1 < NCHUNK) issue_chunk(c + 1, cur ^ 1);

    const float* sx = sx2 + cur * SXHALF;
    const float* sw = sw2 + cur * SWHALF;

    // LN stats: 2 threads per token scan this chunk from LDS
    {
      const int row = tid >> 1;
      const float* p = &sx[row * LDX + (tid & 1) * 32];
      #pragma unroll 8
      for (int j = 0; j < 32; ++j) { const float v = p[j]; psum += v; psq += v * v; }
    }

    // WMMA: M=16 tokens (this wave), N=16, K=BK in steps of 4.
    // A frag (16x4 f32): lanes 0-15 hold K={k,k+1} of row M=lane; lanes 16-31 K={k+2,k+3}.
    // B frag (4x16 f32): symmetric layout on W' rows (N index = lane&15).
    {
      const int r  = lane & 15;
      const int hi = (lane >> 4) << 1;  // 0 or 2
      const float* ap = &sx[(wave * 16 + r) * LDX + hi];
      const float* bp = &sw[r * LDX + hi];
      #pragma unroll
      for (int kk = 0; kk < BK; kk += 4) {
        const v2f a = *(const v2f*)(ap + kk);
        const v2f b = *(const v2f*)(bp + kk);
        acc = __builtin_amdgcn_wmma_f32_16x16x4_f32(false, a, false, b, (short)0, acc,
                                                    false, false);
      }
    }
    cur ^= 1;
  }

  // Finalize LN stats
  spsum[tid] = psum;
  spsq[tid]  = psq;
  __syncthreads();
  if (tid < TOKB) {
    const float s   = spsum[2 * tid] + spsum[2 * tid + 1];
    const float q   = spsq[2 * tid] + spsq[2 * tid + 1];
    const float mu  = s * (1.0f / D);
    const float var = q * (1.0f / D) - mu * mu;
    smu[tid]   = mu;
    srstd[tid] = rsqrtf(var + 1e-5f);
  }
  __syncthreads();

  // Fix up logits: logit = rstd*(proj - mu*S1[n]) + S2[n]; stash in LDS (buf-1 reuse).
  // C/D layout: VGPR r -> M = r (lanes 0-15) or r+8 (lanes 16-31), N = lane&15.
  float* slog = sx2 + SXHALF;
  {
    const int n    = lane & 15;
    const int mofs = (lane >> 4) * 8;
    const float s1 = params[n];
    const float s2 = params[16 + n];
    #pragma unroll
    for (int r = 0; r < 8; ++r) {
      const int ml = wave * 16 + mofs + r;
      slog[ml * 16 + n] = srstd[ml] * (acc[r] - smu[ml] * s1) + s2;
    }
  }
  __syncthreads();

  // Per-token epilogue: one thread per token.
  float* shex = sx2;   // 64 x 64 (buf-0 reuse)
  float* sgw  = sw2;   // 64 x 3
  const float itemp = 1.0f / params[32];
  const float alpha = params[33];

  if (tid < TOKB) {
    const int m = tokBase + tid;
    float l[11];
    #pragma unroll
    for (int j = 0; j < 11; ++j) l[j] = slog[tid * 16 + j];

    // ---- Q2: 4 vertices -> fixed group map [2,1,1,0]
    const float t0 = tanhf(l[0]), t1 = tanhf(l[1]);
    float s2v[4]; float mx = -1e30f;
    #pragma unroll
    for (int v = 0; v < 4; ++v) {
      const float d = ((v & 2) ? t0 : -t0) + ((v & 1) ? t1 : -t1);
      s2v[v] = d * itemp;
      mx = fmaxf(mx, s2v[v]);
    }
    float es = 0.f;
    #pragma unroll
    for (int v = 0; v < 4; ++v) { s2v[v] = expf(s2v[v] - mx); es += s2v[v]; }
    float inv = 1.0f / es;
    float scq2[3];
    scq2[0] = s2v[3] * inv;
    scq2[1] = (s2v[1] + s2v[2]) * inv;
    scq2[2] = s2v[0] * inv;

    // ---- Q3: 8 vertices -> learned (pre-softmaxed) group map
    const float u0 = tanhf(l[2]), u1 = tanhf(l[3]), u2 = tanhf(l[4]);
    float s3v[8]; mx = -1e30f;
    #pragma unroll
    for (int v = 0; v < 8; ++v) {
      const float d = ((v & 4) ? u0 : -u0) + ((v & 2) ? u1 : -u1) + ((v & 1) ? u2 : -u2);
      s3v[v] = d * itemp;
      mx = fmaxf(mx, s3v[v]);
    }
    es = 0.f;
    #pragma unroll
    for (int v = 0; v < 8; ++v) { s3v[v] = expf(s3v[v] - mx); es += s3v[v]; }
    inv = 1.0f / es;
    float scq3[3] = {0.f, 0.f, 0.f};
    #pragma unroll
    for (int v = 0; v < 8; ++v) {
      const float w = s3v[v] * inv;
      scq3[0] += w * params[40 + v * 3 + 0];
      scq3[1] += w * params[40 + v * 3 + 1];
      scq3[2] += w * params[40 + v * 3 + 2];
    }

    // ---- Q6: 64 hexagram vertices (column d = bit (5-d), MSB-first)
    float h6[6];
    #pragma unroll
    for (int j = 0; j < 6; ++j) h6[j] = tanhf(l[5 + j]);
    mx = -1e30f;
    for (int v = 0; v < 64; ++v) {
      float d = 0.f;
      #pragma unroll
      for (int j = 0; j < 6; ++j) d += ((v >> (5 - j)) & 1) ? h6[j] : -h6[j];
      mx = fmaxf(mx, d * itemp);
    }
    es = 0.f;
    for (int v = 0; v < 64; ++v) {
      float d = 0.f;
      #pragma unroll
      for (int j = 0; j < 6; ++j) d += ((v >> (5 - j)) & 1) ? h6[j] : -h6[j];
      es += expf(d * itemp - mx);
    }
    inv = 1.0f / es;
    float hd[6]  = {0.f, 0.f, 0.f, 0.f, 0.f, 0.f};
    float wht[3] = {0.f, 0.f, 0.f};
    for (int v = 0; v < 64; ++v) {
      float d = 0.f;
      #pragma unroll
      for (int j = 0; j < 6; ++j) d += ((v >> (5 - j)) & 1) ? h6[j] : -h6[j];
      const float hw = expf(d * itemp - mx) * inv;
      shex[tid * 64 + v] = hw;
      out[HEX_BASE + (size_t)m * 64 + v] = hw;
      #pragma unroll
      for (int j = 0; j < 6; ++j) hd[j] += ((v >> (5 - j)) & 1) ? hw : -hw;
      const int pc = __popc(v);
      if (pc >= 5) wht[0] += hw;
      if (pc == 3) wht[1] += hw;
      if (pc <= 1) wht[2] += hw;
    }
    wht[0] *= (1.0f / 7.0f);
    wht[1] *= (1.0f / 20.0f);
    wht[2] *= (1.0f / 7.0f);

    // Normalized anchors (compile-time constants from hexagram means)
    const float r5 = 0.44721359549995794f; // 1/sqrt(5)
    float scc[3];
    scc[0] =  r5 * (hd[0] + hd[1] + hd[2] + hd[3] + hd[4]);
    scc[1] = 0.5f * (-hd[0] + hd[1] - hd[2] + hd[5]);
    scc[2] = -r5 * (hd[0] + hd[1] + hd[3] + hd[4] + hd[5]);

    float cb[3]; mx = -1e30f;
    #pragma unroll
    for (int g = 0; g < 3; ++g) {
      const float scq6 = (1.0f - alpha) * scc[g] + alpha * wht[g];
      cb[g] = params[34] * scq2[g] + params[35] * scq3[g] + params[36] * scq6;
      mx = fmaxf(mx, cb[g]);
    }
    es = 0.f;
    #pragma unroll
    for (int g = 0; g < 3; ++g) { cb[g] = expf(cb[g] - mx); es += cb[g]; }
    inv = 1.0f / es;
    #pragma unroll
    for (int g = 0; g < 3; ++g) {
      const float gw = cb[g] * inv;
      out[(size_t)m * 3 + g] = gw;
      sgw[tid * 3 + g] = gw;
    }
  }
  __syncthreads();

  // Deterministic per-block partial sums (fixed-order, no atomics): 64 hex + 3 group.
  if (tid < 64) {
    float s = 0.f;
    for (int t = 0; t < TOKB; ++t) s += shex[t * 64 + tid];
    part[blk * 68 + tid] = s;
  } else if (tid < 67) {
    const int g = tid - 64;
    float s = 0.f;
    for (int t = 0; t < TOKB; ++t) s += sgw[t * 3 + g];
    part[blk * 68 + 64 + g] = s;
  }
}

// ---------------------------------------------------------------------------
// Kernel 3: reduce per-block partials (fixed order) -> lb_loss scalar.
// ---------------------------------------------------------------------------
__global__ void router_finalize(const float* __restrict__ part, float* __restrict__ out) {
  __shared__ float red[68];
  const int tid = threadIdx.x;
  if (tid < 67) {
    float s = 0.f;
    for (int b = 0; b < NBLK; ++b) s += part[b * 68 + tid];
    red[tid] = s * (1.0f / NTOK);
  }
  __syncthreads();
  if (tid == 0) {
    float lb = 0.f;
    for (int g = 0; g < 3; ++g) { const float mg = red[64 + g]; lb += mg * logf(mg + 1e-8f); }
    float lh = 0.f;
    for (int v = 0; v < 64; ++v) { const float mh = red[v]; lh += mh * logf(mh + 1e-8f); }
    out[LB_IDX] = lb + 0.1f * lh;
  }
}

// ---------------------------------------------------------------------------
extern "C" void kernel_launch(void* const* d_in, const int* in_sizes, int n_in,
                              void* d_out, int out_size, void* d_ws, size_t ws_size,
                              hipStream_t stream) {
  const float* x             = (const float*)d_in[0];
  const float* ln_w          = (const float*)d_in[1];
  const float* ln_b          = (const float*)d_in[2];
  const float* w_q2          = (const float*)d_in[3];
  const float* w_q3          = (const float*)d_in[4];
  const float* w_q6          = (const float*)d_in[5];
  const float* log_temp      = (const float*)d_in[6];
  const float* log_scale_mix = (const float*)d_in[7];
  const float* q3_to_group   = (const float*)d_in[8];
  const float* log_wht_mix   = (const float*)d_in[9];

  float* ws     = (float*)d_ws;
  float* Wp     = ws;                 // 16*2048 floats
  float* params = ws + NPAD * D;      // 64 floats
  float* part   = params + 64;        // NBLK*68 floats
  float* out    = (float*)d_out;

  prep_kernel<<<1, 256, 0, stream>>>(ln_w, ln_b, w_q2, w_q3, w_q6, log_temp,
                                     log_scale_mix, q3_to_group, log_wht_mix, Wp, params);
  router_main<<<NBLK, TPB, 0, stream>>>(x, Wp, params, out, part);
  router_finalize<<<1, 128, 0, stream>>>(part, out);
}